// PredictiveGNN_86938728005989
// MI455X (gfx1250) — compile-verified
//
#include <hip/hip_runtime.h>
#include <hip/hip_bf16.h>

#define NN   100000
#define EE   1600000
#define DIN  384
#define HDIM 128
#define HHD  64

typedef __attribute__((ext_vector_type(2))) float v2f;
typedef __attribute__((ext_vector_type(4))) float v4f;
typedef __attribute__((ext_vector_type(8))) float v8f;
typedef int v4i __attribute__((vector_size(16)));

// ---- A-tile staging: async global->LDS (CDNA5) with VGPR-copy fallback ----
__device__ __forceinline__ void stage_tile(const float* __restrict__ Ag,
                                           float* As, int tid, int cnt4) {
#if __has_builtin(__builtin_amdgcn_global_load_async_to_lds_b128)
    typedef __attribute__((address_space(1))) v4i* gp_t;
    typedef __attribute__((address_space(3))) v4i* lp_t;
    for (int i = tid; i < cnt4; i += 256) {
        __builtin_amdgcn_global_load_async_to_lds_b128(
            (gp_t)(Ag + (size_t)i * 4),
            (lp_t)(As + (size_t)i * 4),
            0, 0);
    }
#if __has_builtin(__builtin_amdgcn_s_wait_asynccnt)
    __builtin_amdgcn_s_wait_asynccnt(0);
#else
    asm volatile("s_wait_asynccnt 0" ::: "memory");
#endif
#else
    const v4f* __restrict__ Ag4 = (const v4f*)Ag;
    v4f* As4 = (v4f*)As;
    for (int i = tid; i < cnt4; i += 256) As4[i] = Ag4[i];
#endif
}

// ---------------- degree kernels ----------------
__global__ void k_edge_deg(const int* __restrict__ dst, float* __restrict__ deg, int e) {
    int i = blockIdx.x * blockDim.x + threadIdx.x;
    if (i < e) atomicAdd(&deg[dst[i]], 1.0f);
}

__global__ void k_dinv(float* __restrict__ deg, int n) {
    int i = blockIdx.x * blockDim.x + threadIdx.x;
    if (i < n) deg[i] = rsqrtf(deg[i] + 1.0f);
}

// ------- WMMA GEMM (K=384 projection): C = relu(A @ B + bias) ----------
// block = 256 threads = 8 wave32; wave w owns 16x16 tile at columns w*16.
template<int K, int NC, bool RELU>
__global__ __launch_bounds__(256) void k_gemm(const float* __restrict__ A,
                                              const float* __restrict__ B,
                                              const float* __restrict__ bias,
                                              float* __restrict__ C) {
    constexpr int NT   = NC / 16;
    constexpr int MT   = 8 / NT;
    constexpr int ROWS = MT * 16;
    __shared__ float As[ROWS * K];

    const int tid = threadIdx.x;
    const int m0  = blockIdx.x * ROWS;

    stage_tile(A + (size_t)m0 * K, As, tid, ROWS * K / 4);
    __syncthreads();

    const int lane = tid & 31;
    const int w    = tid >> 5;
    const int half = lane >> 4;
    const int r    = lane & 15;
    const int nt   = w % NT;
    const int mt   = w / NT;
    const int n0   = nt * 16;

    v8f c;
    const float bv = bias ? bias[n0 + r] : 0.0f;
#pragma unroll
    for (int j = 0; j < 8; ++j) c[j] = bv;

    const float* __restrict__ as = &As[(mt * 16 + r) * K];
#pragma unroll 4
    for (int k0 = 0; k0 < K; k0 += 4) {
        const int ka = k0 + 2 * half;
        v2f a, b;
        a.x = as[ka];
        a.y = as[ka + 1];
        b.x = B[(size_t)ka * NC + n0 + r];
        b.y = B[(size_t)(ka + 1) * NC + n0 + r];
        c = __builtin_amdgcn_wmma_f32_16x16x4_f32(false, a, false, b, (short)0, c, false, false);
    }

    const int row0 = m0 + mt * 16 + 8 * half;
#pragma unroll
    for (int j = 0; j < 8; ++j) {
        float v = c[j];
        if (RELU) v = fmaxf(v, 0.0f);
        C[(size_t)(row0 + j) * NC + n0 + r] = v;
    }
}

// ------- WMMA GEMM, K=128, register-resident B panel --------------------
// Each wave preloads its 16 B-columns over all K into 64 VGPRs, then
// grid-strides over M tiles: inner loop is LDS reads + WMMA only.
template<int NC, bool RELU>
__global__ __launch_bounds__(256) void k_gemm128(const float* __restrict__ A,
                                                 const float* __restrict__ B,
                                                 const float* __restrict__ bias,
                                                 float* __restrict__ C,
                                                 int mtiles) {
    constexpr int K    = 128;
    constexpr int NT   = NC / 16;
    constexpr int MT   = 8 / NT;
    constexpr int ROWS = MT * 16;
    __shared__ float As[ROWS * K];

    const int tid  = threadIdx.x;
    const int lane = tid & 31;
    const int w    = tid >> 5;
    const int half = lane >> 4;
    const int r    = lane & 15;
    const int nt   = w % NT;
    const int mt   = w / NT;
    const int n0   = nt * 16;

    // B panel -> registers (64 VGPRs/lane)
    v2f breg[K / 4];
#pragma unroll
    for (int k = 0; k < K / 4; ++k) {
        const int ka = 4 * k + 2 * half;
        breg[k].x = B[(size_t)ka * NC + n0 + r];
        breg[k].y = B[(size_t)(ka + 1) * NC + n0 + r];
    }
    const float bv = bias ? bias[n0 + r] : 0.0f;

    for (int t = blockIdx.x; t < mtiles; t += gridDim.x) {
        const int m0 = t * ROWS;
        __syncthreads();                       // previous iteration done with As
        stage_tile(A + (size_t)m0 * K, As, tid, ROWS * K / 4);
        __syncthreads();

        v8f c;
#pragma unroll
        for (int j = 0; j < 8; ++j) c[j] = bv;

        const float* __restrict__ as = &As[(mt * 16 + r) * K];
#pragma unroll 8
        for (int k = 0; k < K / 4; ++k) {
            const int ka = 4 * k + 2 * half;
            v2f a;
            a.x = as[ka];
            a.y = as[ka + 1];
            c = __builtin_amdgcn_wmma_f32_16x16x4_f32(false, a, false, breg[k],
                                                      (short)0, c, false, false);
        }

        const int row0 = m0 + mt * 16 + 8 * half;
#pragma unroll
        for (int j = 0; j < 8; ++j) {
            float v = c[j];
            if (RELU) v = fmaxf(v, 0.0f);
            C[(size_t)(row0 + j) * NC + n0 + r] = v;
        }
    }
}

// ------- edge aggregation: agg[dst] += hw[src] * dinv[src]*dinv[dst] -------
__global__ void k_agg(const int* __restrict__ src, const int* __restrict__ dst,
                      const float* __restrict__ dinv, const float* __restrict__ hw,
                      float* __restrict__ agg, int e) {
    const int tid  = threadIdx.x;
    const int q    = tid & 31;
    const int sub  = tid >> 5;
    const int eidx = blockIdx.x * 8 + sub;
    if (eidx >= e) return;
    const int s = src[eidx];
    const int d = dst[eidx];
    const float wgt = dinv[s] * dinv[d];
    const v4f vv = ((const v4f*)(hw + (size_t)s * HDIM))[q];
    float* ap = agg + (size_t)d * HDIM + q * 4;
    atomicAdd(ap + 0, vv.x * wgt);
    atomicAdd(ap + 1, vv.y * wgt);
    atomicAdd(ap + 2, vv.z * wgt);
    atomicAdd(ap + 3, vv.w * wgt);
}

// ------- fused GCN bias + BatchNorm(eval) + ReLU + residual -------
__global__ void k_post(const float* __restrict__ agg, const float* __restrict__ hw,
                       const float* __restrict__ hprev, const float* __restrict__ dinv,
                       const float* __restrict__ bc, const float* __restrict__ gamma,
                       const float* __restrict__ beta, const float* __restrict__ rmean,
                       const float* __restrict__ rvar, float* __restrict__ hout,
                       int useRes) {
    const int idx = blockIdx.x * blockDim.x + threadIdx.x;
    if (idx >= NN * HDIM) return;
    const int n = idx >> 7;
    const int f = idx & 127;
    float sn = dinv[n];
    sn *= sn;
    float v = agg[idx] + hw[idx] * sn + bc[f];
    const float scale = gamma[f] * rsqrtf(rvar[f] + 1e-5f);
    v = (v - rmean[f]) * scale + beta[f];
    v = fmaxf(v, 0.0f);
    if (useRes) v += hprev[idx];
    hout[idx] = v;
}

// ------- head second layers: sigmoid(dot64), 4-logit dot64, sigmoid(dot64) ---
__global__ void k_heads(const float* __restrict__ t_rs, const float* __restrict__ t_rl,
                        const float* __restrict__ t_rel,
                        const float* __restrict__ rs_W2, const float* __restrict__ rs_b2,
                        const float* __restrict__ rl_W2, const float* __restrict__ rl_b2,
                        const float* __restrict__ rel_W2, const float* __restrict__ rel_b2,
                        float* __restrict__ out) {
    const int n = blockIdx.x * blockDim.x + threadIdx.x;
    if (n >= NN) return;
    float srs = rs_b2[0];
    float srel = rel_b2[0];
    float l0 = rl_b2[0], l1 = rl_b2[1], l2 = rl_b2[2], l3 = rl_b2[3];
#pragma unroll 8
    for (int k = 0; k < HHD; ++k) {
        const float a = t_rs[(size_t)n * HHD + k];
        srs += a * rs_W2[k];
        const float b = t_rel[(size_t)n * HHD + k];
        srel += b * rel_W2[k];
        const float c = t_rl[(size_t)n * HHD + k];
        l0 += c * rl_W2[k * 4 + 0];
        l1 += c * rl_W2[k * 4 + 1];
        l2 += c * rl_W2[k * 4 + 2];
        l3 += c * rl_W2[k * 4 + 3];
    }
    out[n] = 1.0f / (1.0f + __expf(-srs));
    float* lo = out + NN + (size_t)n * 4;
    lo[0] = l0; lo[1] = l1; lo[2] = l2; lo[3] = l3;
    out[(size_t)5 * NN + n] = 1.0f / (1.0f + __expf(-srel));
}

extern "C" void kernel_launch(void* const* d_in, const int* in_sizes, int n_in,
                              void* d_out, int out_size, void* d_ws, size_t ws_size,
                              hipStream_t stream) {
    (void)in_sizes; (void)n_in; (void)out_size; (void)ws_size;
    const float* x      = (const float*)d_in[0];
    const int*   ei     = (const int*)d_in[1];
    const float* Wp     = (const float*)d_in[2];
    const float* bp     = (const float*)d_in[3];
    const float* Wc     = (const float*)d_in[4];
    const float* bc     = (const float*)d_in[5];
    const float* gamma  = (const float*)d_in[6];
    const float* beta   = (const float*)d_in[7];
    const float* rmean  = (const float*)d_in[8];
    const float* rvar   = (const float*)d_in[9];
    const float* rel_W1 = (const float*)d_in[10];
    const float* rel_b1 = (const float*)d_in[11];
    const float* rel_W2 = (const float*)d_in[12];
    const float* rel_b2 = (const float*)d_in[13];
    const float* rs_W1  = (const float*)d_in[14];
    const float* rs_b1  = (const float*)d_in[15];
    const float* rs_W2  = (const float*)d_in[16];
    const float* rs_b2  = (const float*)d_in[17];
    const float* rl_W1  = (const float*)d_in[18];
    const float* rl_b1  = (const float*)d_in[19];
    const float* rl_W2  = (const float*)d_in[20];
    const float* rl_b2  = (const float*)d_in[21];

    const int* srcI = ei;
    const int* dstI = ei + EE;
    float* out = (float*)d_out;

    // workspace layout (floats): dinv | bufA | bufB | hw | agg  (~205 MB)
    float* ws   = (float*)d_ws;
    float* dinv = ws;
    float* bufA = ws + 102400;
    float* bufB = bufA + (size_t)NN * HDIM;
    float* hw   = bufB + (size_t)NN * HDIM;
    float* agg  = hw   + (size_t)NN * HDIM;

    // degrees -> dinv
    (void)hipMemsetAsync(dinv, 0, NN * sizeof(float), stream);
    k_edge_deg<<<(EE + 255) / 256, 256, 0, stream>>>(dstI, dinv, EE);
    k_dinv<<<(NN + 255) / 256, 256, 0, stream>>>(dinv, NN);

    // projection: h0 = relu(x @ Wp + bp)
    k_gemm<DIN, HDIM, true><<<NN / 16, 256, 0, stream>>>(x, Wp, bp, bufA);

    float* hin = bufA;
    float* hout = bufB;
    for (int i = 0; i < 3; ++i) {
        k_gemm128<HDIM, false><<<2048, 256, 0, stream>>>(
            hin, Wc + (size_t)i * HDIM * HDIM, nullptr, hw, NN / 16);
        (void)hipMemsetAsync(agg, 0, (size_t)NN * HDIM * sizeof(float), stream);
        k_agg<<<(EE + 7) / 8, 256, 0, stream>>>(srcI, dstI, dinv, hw, agg, EE);
        k_post<<<(NN * HDIM + 255) / 256, 256, 0, stream>>>(
            agg, hw, hin, dinv,
            bc + i * HDIM, gamma + i * HDIM, beta + i * HDIM,
            rmean + i * HDIM, rvar + i * HDIM, hout, i > 0 ? 1 : 0);
        float* t = hin; hin = hout; hout = t;
    }
    float* hfin  = hin;                        // = bufB after 3 swaps
    float* t_rs  = hw;
    float* t_rl  = agg;
    float* t_rel = (hfin == bufB) ? bufA : bufB;

    k_gemm128<HHD, true><<<2048, 256, 0, stream>>>(hfin, rs_W1, rs_b1, t_rs, NN / 32);
    k_gemm128<HHD, true><<<2048, 256, 0, stream>>>(hfin, rl_W1, rl_b1, t_rl, NN / 32);
    k_gemm128<HHD, true><<<2048, 256, 0, stream>>>(hfin, rel_W1, rel_b1, t_rel, NN / 32);

    k_heads<<<(NN + 255) / 256, 256, 0, stream>>>(t_rs, t_rl, t_rel,
                                                  rs_W2, rs_b2, rl_W2, rl_b2,
                                                  rel_W2, rel_b2, out);
}